// Deformation4D_7687991460415
// MI455X (gfx1250) — compile-verified
//
#include <hip/hip_runtime.h>
#include <cstdint>

#define K_TOT 300
#define K_NB  10
#define BLOCK 256

// ---------------------------------------------------------------------------
// Tensor Data Mover staging of the control-point tables into LDS (gfx1250).
// ---------------------------------------------------------------------------
#if defined(__HIP_DEVICE_COMPILE__) && defined(__gfx1250__) && __has_builtin(__builtin_amdgcn_tensor_load_to_lds)
#define USE_TDM 1
#else
#define USE_TDM 0
#endif

#if USE_TDM
typedef __attribute__((ext_vector_type(4))) unsigned int u32x4;
typedef __attribute__((ext_vector_type(8))) int          i32x8;
typedef __attribute__((ext_vector_type(4))) int          i32x4;

// One TDM descriptor copying `ndw` DWORDs (ndw < 65536) global -> LDS.
// D# layout per CDNA5 ISA (degenerate 2-D copy).
__device__ __forceinline__ void tdm_copy_1d(uint32_t lds_byte_addr,
                                            const void* gsrc, uint32_t ndw) {
  const uint64_t ga = (uint64_t)(uintptr_t)gsrc;
  u32x4 g0;
  g0.x = 1u;                                                 // count=1, user D#
  g0.y = lds_byte_addr;                                      // lds_addr
  g0.z = (uint32_t)ga;                                       // global_addr[31:0]
  g0.w = (uint32_t)((ga >> 32) & 0x1FFFFFFull) | (2u << 30); // addr[56:32]|type=2
  i32x8 g1;
  g1[0] = (int)(2u << 16);                  // wg_mask=0, data_size=2 (4 bytes)
  g1[1] = (int)((ndw & 0xFFFFu) << 16);     // tensor_dim0[15:0]
  g1[2] = (int)((ndw >> 16) | (1u << 16));  // tensor_dim0[31:16] | tensor_dim1=1
  g1[3] = (int)((ndw & 0xFFFFu) << 16);     // tile_dim0 = ndw
  g1[4] = 1;                                // tile_dim1=1, tile_dim2=0 (unused)
  g1[5] = (int)ndw;                         // tensor_dim0_stride[31:0]
  g1[6] = (int)((ndw & 0xFFFFu) << 16);     // stride0 hi=0 | dim1_stride[15:0]
  g1[7] = (int)(ndw >> 16);                 // dim1_stride[47:16]
  i32x4 gz = {0, 0, 0, 0};
#if __clang_major__ >= 23
  i32x8 gz8 = {0, 0, 0, 0, 0, 0, 0, 0};
  __builtin_amdgcn_tensor_load_to_lds(g0, g1, gz, gz, gz8, 0);
#else
  __builtin_amdgcn_tensor_load_to_lds(g0, g1, gz, gz, 0);
#endif
}
#endif  // USE_TDM

// ---------------------------------------------------------------------------
// Deformation kernel: one thread per gaussian.
// Per block: TDM-stage raw tables, precompute per-control-point affine
// transforms (R, b = p + t - R*p) once, then deformed = R*m + b per neighbor.
// ---------------------------------------------------------------------------
__global__ __launch_bounds__(BLOCK) void Deformation4D_kernel(
    const float* __restrict__ means,     // [N,3]
    const float* __restrict__ quats,     // [N,4] (w,x,y,z)
    const float* __restrict__ weights,   // [N,10]
    const float* __restrict__ ctrans,    // [300,3]
    const float* __restrict__ crots,     // [300,4]
    const float* __restrict__ cpos,      // [300,3]
    const int*   __restrict__ indices,   // [N,10]
    float* __restrict__ out_means,       // [N,3]
    float* __restrict__ out_quats,       // [N,4]
    int n) {
  __shared__ __align__(16) float s_rot[K_TOT * 4];    // raw quats, 4800 B
  __shared__ __align__(16) float s_aff[K_TOT * 12];   // [R|b] rows,  14400 B
  __shared__ __align__(16) float s_stage[K_TOT * 6];  // trans[900] pos[900], 7200 B

#if USE_TDM
  if (threadIdx.x == 0) {
    tdm_copy_1d((uint32_t)(uintptr_t)s_rot,                     crots,  K_TOT * 4);
    tdm_copy_1d((uint32_t)(uintptr_t)s_stage,                   ctrans, K_TOT * 3);
    tdm_copy_1d((uint32_t)(uintptr_t)s_stage + K_TOT * 3 * 4,   cpos,   K_TOT * 3);
    __builtin_amdgcn_s_wait_tensorcnt(0);
  }
  // TDM writes LDS through a descriptor clang can't see; force reload.
  asm volatile("" : : "r"(s_rot), "r"(s_stage) : "memory");
  __syncthreads();
#else
  for (int t = threadIdx.x; t < K_TOT * 4; t += BLOCK) s_rot[t] = crots[t];
  for (int t = threadIdx.x; t < K_TOT * 3; t += BLOCK) {
    s_stage[t]             = ctrans[t];
    s_stage[K_TOT * 3 + t] = cpos[t];
  }
  __syncthreads();
#endif

  const float4* s_rot4 = reinterpret_cast<const float4*>(s_rot);

  // ---- per-block precompute: affine transform per control point ----
  for (int t = threadIdx.x; t < K_TOT; t += BLOCK) {
    const float4 q = s_rot4[t];
    const float qq  = q.x * q.x + q.y * q.y + q.z * q.z + q.w * q.w;
    const float inv = 1.0f / fmaxf(sqrtf(qq), 1e-8f);
    const float w_ = q.x * inv, x_ = q.y * inv, y_ = q.z * inv, z_ = q.w * inv;

    const float xx = x_ * x_, yy = y_ * y_, zz = z_ * z_;
    const float xy = x_ * y_, xz = x_ * z_, yz = y_ * z_;
    const float wx = w_ * x_, wy = w_ * y_, wz = w_ * z_;

    const float r00 = 1.0f - 2.0f * (yy + zz), r01 = 2.0f * (xy - wz), r02 = 2.0f * (xz + wy);
    const float r10 = 2.0f * (xy + wz), r11 = 1.0f - 2.0f * (xx + zz), r12 = 2.0f * (yz - wx);
    const float r20 = 2.0f * (xz - wy), r21 = 2.0f * (yz + wx), r22 = 1.0f - 2.0f * (xx + yy);

    const float tx = s_stage[3 * t + 0], ty = s_stage[3 * t + 1], tz = s_stage[3 * t + 2];
    const float px = s_stage[K_TOT * 3 + 3 * t + 0];
    const float py = s_stage[K_TOT * 3 + 3 * t + 1];
    const float pz = s_stage[K_TOT * 3 + 3 * t + 2];

    const float bx = px + tx - (r00 * px + r01 * py + r02 * pz);
    const float by = py + ty - (r10 * px + r11 * py + r12 * pz);
    const float bz = pz + tz - (r20 * px + r21 * py + r22 * pz);

    float4* arow = reinterpret_cast<float4*>(s_aff + 12 * t);
    arow[0] = make_float4(r00, r01, r02, bx);
    arow[1] = make_float4(r10, r11, r12, by);
    arow[2] = make_float4(r20, r21, r22, bz);
  }
  __syncthreads();

  const int i = blockIdx.x * BLOCK + threadIdx.x;
  if (i >= n) return;

  // ---- per-gaussian loads (coalesced; b128 quat, b64 weight/index rows) ----
  const float* mrow = means + (size_t)3 * i;
  const float mx = mrow[0], my = mrow[1], mz = mrow[2];
  const float4 myq = reinterpret_cast<const float4*>(quats)[i];

  float wk[K_NB];
  int   jk[K_NB];
  const float2* wr = reinterpret_cast<const float2*>(weights + (size_t)K_NB * i);
  const int2*   ir = reinterpret_cast<const int2*>(indices + (size_t)K_NB * i);
#pragma unroll
  for (int t = 0; t < K_NB / 2; ++t) {
    const float2 a = wr[t];
    wk[2 * t] = a.x; wk[2 * t + 1] = a.y;
    const int2 b = ir[t];
    jk[2 * t] = b.x; jk[2 * t + 1] = b.y;
  }

  // ---- blend over 10 neighbors: 4x ds_load_b128 + ~16 FMA each ----
  float amx = 0.f, amy = 0.f, amz = 0.f;
  float aqw = 0.f, aqx = 0.f, aqy = 0.f, aqz = 0.f;
#pragma unroll
  for (int k = 0; k < K_NB; ++k) {
    const int   j = jk[k];
    const float w = wk[k];

    const float4* arow = reinterpret_cast<const float4*>(s_aff + 12 * j);
    const float4 r0 = arow[0], r1 = arow[1], r2 = arow[2];
    const float4 q  = s_rot4[j];

    const float ox = fmaf(r0.x, mx, fmaf(r0.y, my, fmaf(r0.z, mz, r0.w)));
    const float oy = fmaf(r1.x, mx, fmaf(r1.y, my, fmaf(r1.z, mz, r1.w)));
    const float oz = fmaf(r2.x, mx, fmaf(r2.y, my, fmaf(r2.z, mz, r2.w)));

    amx = fmaf(w, ox, amx);
    amy = fmaf(w, oy, amy);
    amz = fmaf(w, oz, amz);
    aqw = fmaf(w, q.x, aqw);
    aqx = fmaf(w, q.y, aqx);
    aqy = fmaf(w, q.z, aqy);
    aqz = fmaf(w, q.w, aqz);
  }

  // ---- normalize blended quaternion, Hamilton product with own quat ----
  const float qn   = sqrtf(aqw * aqw + aqx * aqx + aqy * aqy + aqz * aqz);
  const float qinv = 1.0f / fmaxf(qn, 1e-8f);
  const float bw = aqw * qinv, bx = aqx * qinv, by = aqy * qinv, bz = aqz * qinv;

  float4 oq;
  oq.x = bw * myq.x - bx * myq.y - by * myq.z - bz * myq.w;
  oq.y = bw * myq.y + bx * myq.x + by * myq.w - bz * myq.z;
  oq.z = bw * myq.z - bx * myq.w + by * myq.x + bz * myq.y;
  oq.w = bw * myq.w + bx * myq.z - by * myq.y + bz * myq.x;

  float* om = out_means + (size_t)3 * i;
  om[0] = amx; om[1] = amy; om[2] = amz;
  reinterpret_cast<float4*>(out_quats)[i] = oq;  // 16B-aligned for N%4==0
}

// ---------------------------------------------------------------------------
extern "C" void kernel_launch(void* const* d_in, const int* in_sizes, int n_in,
                              void* d_out, int out_size, void* d_ws, size_t ws_size,
                              hipStream_t stream) {
  const float* means   = (const float*)d_in[0];
  const float* quats   = (const float*)d_in[1];
  const float* weights = (const float*)d_in[2];
  const float* ctrans  = (const float*)d_in[3];
  const float* crots   = (const float*)d_in[4];
  const float* cpos    = (const float*)d_in[5];
  const int*   indices = (const int*)d_in[6];

  const int n = in_sizes[0] / 3;  // N gaussians
  float* out_means = (float*)d_out;
  float* out_quats = out_means + (size_t)3 * n;

  const int blocks = (n + BLOCK - 1) / BLOCK;
  Deformation4D_kernel<<<blocks, BLOCK, 0, stream>>>(
      means, quats, weights, ctrans, crots, cpos, indices,
      out_means, out_quats, n);
}